// GraphModel_37245956391549
// MI455X (gfx1250) — compile-verified
//
#include <hip/hip_runtime.h>
#include <hip/hip_bf16.h>

typedef __attribute__((ext_vector_type(2))) float v2f;
typedef __attribute__((ext_vector_type(8))) float v8f;

#define NNODES 65536
#define NGRAPH 64
#define NPG    1024
#define EPG    16384
#define NEDGE  (NGRAPH * EPG)
#define NHEAD  4
#define DFEAT  32
#define HD     128
#define NEG_SLOPE 0.2f

// ---- monotone float <-> uint encoding for atomicMax on signed floats ----
__device__ __forceinline__ unsigned encF(float f) {
    unsigned u = __float_as_uint(f);
    return (u & 0x80000000u) ? ~u : (u | 0x80000000u);
}
__device__ __forceinline__ float decF(unsigned k) {
    unsigned u = (k & 0x80000000u) ? (k & 0x7fffffffu) : ~k;
    return __uint_as_float(u);
}
#define ENC_NEG_INF 0x007fffffu   // encF(-inf)

// ============================================================
// fp32 WMMA GEMM: out[M,128] = A[M,K] @ W[K,128] + bias[128]
// one wave computes a 16x16 tile with V_WMMA_F32_16X16X4_F32
// A frag 16x4: lanes 0-15 hold K=k,k+1 ; lanes 16-31 hold K=k+2,k+3
// B frag 4x16: mirrored over N
// C/D: 8 VGPRs, lanes 0-15 -> M=r, lanes 16-31 -> M=r+8
// ============================================================
__global__ void gemm_wmma_f32(const float* __restrict__ A,
                              const float* __restrict__ W,
                              const float* __restrict__ bias,
                              float* __restrict__ out,
                              int M, int K) {
    const int wave = threadIdx.x >> 5;     // 8 waves / block
    const int lane = threadIdx.x & 31;
    const int tm = blockIdx.x * 16;        // 16-row strip per block
    const int tn = wave * 16;              // 8 column tiles (N=128)
    if (tm >= M) return;

    const int lh    = lane & 15;
    const int khalf = (lane >> 4) << 1;    // 0 or 2

    v8f acc = {};
    const float* arow = A + (size_t)(tm + lh) * K;
    for (int k = 0; k < K; k += 4) {
        v2f a, b;
        a.x = arow[k + khalf];
        a.y = arow[k + khalf + 1];
        b.x = W[(size_t)(k + khalf)     * HD + (tn + lh)];
        b.y = W[(size_t)(k + khalf + 1) * HD + (tn + lh)];
        acc = __builtin_amdgcn_wmma_f32_16x16x4_f32(
            false, a, false, b, (short)0, acc, false, false);
    }

    const int col   = tn + lh;
    const int rbase = tm + ((lane >> 4) ? 8 : 0);
    const float bv  = bias[col];
#pragma unroll
    for (int r = 0; r < 8; ++r)
        out[(size_t)(rbase + r) * HD + col] = acc[r] + bv;
}

// ============================================================
// init per-layer state: menc=-inf (segment max), hEnc=-inf (head maxpool)
// ============================================================
__global__ void init_layer(unsigned* __restrict__ menc,
                           unsigned* __restrict__ hEnc) {
    int t = blockIdx.x * blockDim.x + threadIdx.x;   // over N*32
    if (t < NNODES * NHEAD) menc[t] = ENC_NEG_INF;
    if (t < NNODES * DFEAT) hEnc[t] = ENC_NEG_INF;
}

// ============================================================
// E1: per-edge logits s[e,h] + segment-max over dst (one wave/edge)
// ============================================================
__global__ void edge_logits_max(const float* __restrict__ fs,
                                const float* __restrict__ fd,
                                const int* __restrict__ src,
                                const int* __restrict__ dst,
                                const float* __restrict__ attn,  // [H*D]
                                float* __restrict__ sbuf,        // [E,H]
                                unsigned* __restrict__ menc) {   // [N,H]
    const int wid  = blockIdx.x * (blockDim.x >> 5) + (threadIdx.x >> 5);
    if (wid >= NEDGE) return;
    const int lane = threadIdx.x & 31;
    const int s = src[wid], d = dst[wid];
    const float* pfs = fs + (size_t)s * HD;
    const float* pfd = fd + (size_t)d * HD;
#pragma unroll
    for (int h = 0; h < NHEAD; ++h) {
        const int idx = h * DFEAT + lane;
        float v = pfs[idx] + pfd[idx];
        v = v > 0.f ? v : NEG_SLOPE * v;              // leaky_relu
        float p = v * attn[idx];
#pragma unroll
        for (int off = 16; off; off >>= 1)
            p += __shfl_xor(p, off, 32);
        if (lane == 0) {
            sbuf[(size_t)wid * NHEAD + h] = p;
            atomicMax(&menc[(size_t)d * NHEAD + h], encF(p));
        }
    }
}

// ============================================================
// E2 (fused exp + normalize + aggregate): one block per (graph, head).
// LDS: numerator [NPG*DFEAT] (128 KB) + denominator [NPG] (4 KB).
//   a      = exp(s[e,h] - m[dst,h])
//   num[n] += a * fs[src, h, :]      (ds_add_f32)
//   den[n] += a
// then out = num/den, head-maxpool via encoded global atomicMax.
// ============================================================
__global__ void edge_aggregate(const float* __restrict__ fs,
                               const float* __restrict__ sbuf,
                               const unsigned* __restrict__ menc,
                               const int* __restrict__ src,
                               const int* __restrict__ dst,
                               unsigned* __restrict__ hEnc) {   // [N,D]
    extern __shared__ float smem[];
    float* num = smem;                    // [NPG * DFEAT]
    float* den = smem + NPG * DFEAT;      // [NPG]
    const int g = blockIdx.x / NHEAD;
    const int h = blockIdx.x % NHEAD;
    const int tid = threadIdx.x;
    for (int i = tid; i < NPG * DFEAT + NPG; i += blockDim.x) smem[i] = 0.f;
    __syncthreads();

    const int wave = tid >> 5, lane = tid & 31;
    const int ebase = g * EPG;
    for (int e = wave; e < EPG; e += 8) {
        const int ge = ebase + e;
        const int s  = src[ge];
        const int dg = dst[ge];
        const float m = decF(menc[(size_t)dg * NHEAD + h]);
        const float a = __expf(sbuf[(size_t)ge * NHEAD + h] - m);
        const int dloc = dg - g * NPG;
        const float val = a * fs[(size_t)s * HD + h * DFEAT + lane];
        atomicAdd(&num[dloc * DFEAT + lane], val);    // ds_add_f32
        if (lane == 0) atomicAdd(&den[dloc], a);      // ds_add_f32
    }
    __syncthreads();

    for (int i = tid; i < NPG * DFEAT; i += blockDim.x) {
        const int nl = i >> 5;
        const float zz = den[nl];
        const float r  = (zz > 0.f) ? (num[i] / zz) : 0.f;  // no-edge node -> 0
        const int n = g * NPG + nl;
        atomicMax(&hEnc[(size_t)n * DFEAT + (i & 31)], encF(r));
    }
}

// ============================================================
// decode encoded maxpool result back to float
// ============================================================
__global__ void decode_enc(const unsigned* __restrict__ hEnc,
                           float* __restrict__ hOut) {
    int t = blockIdx.x * blockDim.x + threadIdx.x;   // over N*D
    if (t < NNODES * DFEAT) hOut[t] = decF(hEnc[t]);
}

// ============================================================
// Global attention pooling: one block per graph
// ============================================================
__global__ void pool_kernel(const float* __restrict__ h2,
                            const float* __restrict__ gW,   // [32]
                            const float* __restrict__ gb,   // [1]
                            float* __restrict__ out) {      // [B,32]
    const int g = blockIdx.x, tid = threadIdx.x;  // 256 threads
    __shared__ float gsh[NPG];
    __shared__ float red[256];
    __shared__ float accw[8][32];

    for (int n = tid; n < NPG; n += 256) {
        const float* p = h2 + (size_t)(g * NPG + n) * DFEAT;
        float a = 0.f;
#pragma unroll
        for (int d = 0; d < DFEAT; ++d) a += p[d] * gW[d];
        gsh[n] = a + gb[0];
    }
    __syncthreads();

    float mx = -3.402823466e38f;
    for (int n = tid; n < NPG; n += 256) mx = fmaxf(mx, gsh[n]);
    red[tid] = mx; __syncthreads();
    for (int s = 128; s; s >>= 1) {
        if (tid < s) red[tid] = fmaxf(red[tid], red[tid + s]);
        __syncthreads();
    }
    const float gm = red[0]; __syncthreads();

    float sm = 0.f;
    for (int n = tid; n < NPG; n += 256) sm += __expf(gsh[n] - gm);
    red[tid] = sm; __syncthreads();
    for (int s = 128; s; s >>= 1) {
        if (tid < s) red[tid] += red[tid + s];
        __syncthreads();
    }
    const float gz = red[0]; __syncthreads();

    const int wave = tid >> 5, lane = tid & 31;
    float part = 0.f;
    for (int n = wave; n < NPG; n += 8) {
        const float w = __expf(gsh[n] - gm) / gz;
        part += w * h2[(size_t)(g * NPG + n) * DFEAT + lane];
    }
    accw[wave][lane] = part;
    __syncthreads();
    if (tid < 32) {
        float t = 0.f;
#pragma unroll
        for (int w = 0; w < 8; ++w) t += accw[w][tid];
        out[g * DFEAT + tid] = t;
    }
}

// ============================================================
extern "C" void kernel_launch(void* const* d_in, const int* in_sizes, int n_in,
                              void* d_out, int out_size, void* d_ws, size_t ws_size,
                              hipStream_t stream) {
    const float* x     = (const float*)d_in[0];
    const int*   src   = (const int*)  d_in[1];
    const int*   dst   = (const int*)  d_in[2];
    // d_in[3] = gid (unused: graphs are contiguous blocks of 1024 nodes)
    const float* Ws1   = (const float*)d_in[4];
    const float* bs1   = (const float*)d_in[5];
    const float* Wd1   = (const float*)d_in[6];
    const float* bd1   = (const float*)d_in[7];
    const float* attn1 = (const float*)d_in[8];
    const float* Ws2   = (const float*)d_in[9];
    const float* bs2   = (const float*)d_in[10];
    const float* Wd2   = (const float*)d_in[11];
    const float* bd2   = (const float*)d_in[12];
    const float* attn2 = (const float*)d_in[13];
    const float* gW    = (const float*)d_in[14];
    const float* gb    = (const float*)d_in[15];
    float* out = (float*)d_out;

    // workspace carve-up (floats) -- ~105 MB total
    float*    ws   = (float*)d_ws;
    float*    fs   = ws;                                   // N*128
    float*    fd   = fs   + (size_t)NNODES * HD;           // N*128
    float*    sbuf = fd   + (size_t)NNODES * HD;           // E*H
    unsigned* menc = (unsigned*)(sbuf + (size_t)NEDGE * NHEAD); // N*H
    unsigned* hEnc = menc + (size_t)NNODES * NHEAD;             // N*D
    float*    h1   = (float*)(hEnc + (size_t)NNODES * DFEAT);   // N*D
    float*    h2   = h1 + (size_t)NNODES * DFEAT;               // N*D

    const dim3 blk(256);
    const int gemmGrid = NNODES / 16;            // 4096 blocks, 8 waves/block
    const int edgeGrid = NEDGE / 8;              // one wave per edge
    const int ndGrid   = (NNODES * DFEAT) / 256;
    const size_t ldsBytes = (size_t)(NPG * DFEAT + NPG) * sizeof(float); // 132 KB

    // -------- layer 1 --------
    init_layer<<<ndGrid, blk, 0, stream>>>(menc, hEnc);
    gemm_wmma_f32<<<gemmGrid, blk, 0, stream>>>(x, Ws1, bs1, fs, NNODES, 128);
    gemm_wmma_f32<<<gemmGrid, blk, 0, stream>>>(x, Wd1, bd1, fd, NNODES, 128);
    edge_logits_max<<<edgeGrid, blk, 0, stream>>>(fs, fd, src, dst, attn1, sbuf, menc);
    edge_aggregate<<<NGRAPH * NHEAD, blk, ldsBytes, stream>>>(fs, sbuf, menc, src, dst, hEnc);
    decode_enc<<<ndGrid, blk, 0, stream>>>(hEnc, h1);

    // -------- layer 2 --------
    init_layer<<<ndGrid, blk, 0, stream>>>(menc, hEnc);
    gemm_wmma_f32<<<gemmGrid, blk, 0, stream>>>(h1, Ws2, bs2, fs, NNODES, 32);
    gemm_wmma_f32<<<gemmGrid, blk, 0, stream>>>(h1, Wd2, bd2, fd, NNODES, 32);
    edge_logits_max<<<edgeGrid, blk, 0, stream>>>(fs, fd, src, dst, attn2, sbuf, menc);
    edge_aggregate<<<NGRAPH * NHEAD, blk, ldsBytes, stream>>>(fs, sbuf, menc, src, dst, hEnc);
    decode_enc<<<ndGrid, blk, 0, stream>>>(hEnc, h2);

    // -------- global attention pooling --------
    pool_kernel<<<NGRAPH, blk, 0, stream>>>(h2, gW, gb, out);
}